// GROUPED_GEMM_51642686767910
// MI455X (gfx1250) — compile-verified
//
#include <hip/hip_runtime.h>

typedef __attribute__((ext_vector_type(2))) float v2f;
typedef __attribute__((ext_vector_type(4))) float v4f;
typedef __attribute__((ext_vector_type(8))) float v8f;

#define GCNT   64
#define KDIM   256
#define NDIM   256
#define SMAX   16384
#define BM     128
#define BN     256
#define BK     16
#define LDA    28            // A tile K-stride (floats): 112B, 16B-multiple, conflict-free rows
#define LDB    136           // B half-tile row stride (floats): 544B <= 1020B 2addr reach, 16B-mult
#define MTILES (SMAX / BM)   // 128 row tiles per group (upper bound)

#if __has_builtin(__builtin_amdgcn_global_load_async_to_lds_b128) && \
    __has_builtin(__builtin_amdgcn_s_wait_asynccnt)
#define USE_ASYNC 1
#else
#define USE_ASYNC 0
#endif

#if USE_ASYNC
// Parameter types per hipcc diagnostic: AS(1) int4* (global src), AS(3) int4* (LDS dst)
typedef int i4 __attribute__((vector_size(4 * sizeof(int))));
typedef __attribute__((address_space(1))) i4 gi4_t;
typedef __attribute__((address_space(3))) i4 li4_t;

__device__ __forceinline__ gi4_t* as_global_b128(const void* p) {
    return (gi4_t*)(unsigned long long)p;
}
__device__ __forceinline__ li4_t* as_lds_b128(void* p) {
    // low 32 bits of a generic LDS pointer are the LDS byte offset
    return (li4_t*)(unsigned int)(unsigned long long)p;
}
#endif

__device__ __forceinline__ void wait_stage() {
#if USE_ASYNC
    __builtin_amdgcn_s_wait_asynccnt(0);   // this wave's async->LDS transfers done
#endif
    __syncthreads();                       // all waves' transfers visible
}

__global__ __launch_bounds__(256)
void GROUPED_GEMM_wmma_f32(const float* __restrict__ A,
                           const float* __restrict__ B,
                           const long long* __restrict__ batch_sizes,
                           const long long* __restrict__ batch_offsets,
                           float* __restrict__ out)
{
    __shared__ float lA[2][BM * LDA];        // 2 x 14336 B
    __shared__ float lB[2][2][BK * LDB];     // 2 bufs x 2 column-halves x 8704 B  (total 62 KB)

    const int g  = blockIdx.x / MTILES;
    const int mt = blockIdx.x % MTILES;

    const long long size = batch_sizes[g];
    const long long row0 = (long long)mt * BM;
    if (row0 >= size) return;                 // uniform early-exit (EXEC stays full)
    const long long off = batch_offsets[g];

    const int tid  = threadIdx.x;
    const int lane = tid & 31;
    const int wave = tid >> 5;
    const int wn   = wave & 3;                // 4 waves along N (64 cols each)
    const int wm   = wave >> 2;               // 2 waves along M (64 rows each)
    const int lm   = lane & 15;
    const int kh   = lane >> 4;               // K half-select for frags / M half for C

    const float* Bg = B + (long long)g * KDIM * NDIM;

    // ---- stage one 16-deep K chunk (A: 128x16, B: 16x256) into LDS buffer `buf` ----
    auto stage = [&](int buf, int k0) {
        #pragma unroll
        for (int i = 0; i < 2; ++i) {                 // A: 512 quads / 256 threads
            const int qid = i * 256 + tid;
            const int r   = qid >> 2;                 // 0..127
            const int q   = (qid & 3) * 4;            // 0,4,8,12
            const long long grow = row0 + r;
            const long long arow = off + ((grow < size) ? grow : 0);  // clamp ragged tail
            const float* gsrc = A + arow * KDIM + k0 + q;
            float*       ldst = &lA[buf][r * LDA + q];
#if USE_ASYNC
            __builtin_amdgcn_global_load_async_to_lds_b128(
                as_global_b128(gsrc), as_lds_b128(ldst), 0, 0);
#else
            *reinterpret_cast<v4f*>(ldst) = *reinterpret_cast<const v4f*>(gsrc);
#endif
        }
        #pragma unroll
        for (int i = 0; i < 4; ++i) {                 // B: 1024 quads / 256 threads
            const int qid  = i * 256 + tid;
            const int kr   = qid >> 6;                // 0..15
            const int q    = (qid & 63) * 4;          // 0..252
            const int half = q >> 7;                  // column half 0/1
            const int qh   = q & 127;                 // 0..124 within half
            const float* gsrc = Bg + (long long)(k0 + kr) * NDIM + q;
            float*       ldst = &lB[buf][half][kr * LDB + qh];
#if USE_ASYNC
            __builtin_amdgcn_global_load_async_to_lds_b128(
                as_global_b128(gsrc), as_lds_b128(ldst), 0, 0);
#else
            *reinterpret_cast<v4f*>(ldst) = *reinterpret_cast<const v4f*>(gsrc);
#endif
        }
    };

    v8f acc[4][4];
    #pragma unroll
    for (int mi = 0; mi < 4; ++mi)
        #pragma unroll
        for (int ni = 0; ni < 4; ++ni)
            acc[mi][ni] = (v8f){0.f, 0.f, 0.f, 0.f, 0.f, 0.f, 0.f, 0.f};

    const int bhalf = wn >> 1;                 // which B column-half this wave reads
    const int bcol0 = (wn & 1) * 64 + lm;      // base column within the half

    // ---- compute one staged chunk out of LDS buffer `buf` (compile-time) ----
    auto compute = [&](int buf) {
        #pragma unroll
        for (int kk = 0; kk < BK; kk += 4) {
            v2f afrag[4], bfrag[4];
            // A 16x4 frag: lanes0-15 -> K=kk+{0,1}; lanes16-31 -> K=kk+{2,3}
            #pragma unroll
            for (int mi = 0; mi < 4; ++mi)
                afrag[mi] = *reinterpret_cast<const v2f*>(
                    &lA[buf][(wm * 64 + mi * 16 + lm) * LDA + kk + 2 * kh]);
            // B 4x16 frag: {k, k+1} pair = one ds_load_2addr_b32 (stride LDB dwords)
            #pragma unroll
            for (int ni = 0; ni < 4; ++ni) {
                const float* bp = &lB[buf][bhalf][(kk + 2 * kh) * LDB + bcol0 + ni * 16];
                bfrag[ni] = (v2f){ bp[0], bp[LDB] };
            }
            #pragma unroll
            for (int mi = 0; mi < 4; ++mi)
                #pragma unroll
                for (int ni = 0; ni < 4; ++ni)
                    acc[mi][ni] = __builtin_amdgcn_wmma_f32_16x16x4_f32(
                        false, afrag[mi], false, bfrag[ni],
                        (short)0, acc[mi][ni], false, false);
        }
    };

    stage(0, 0);
    wait_stage();

    // two chunks per iteration -> compile-time LDS buffer indices, 8 iterations
    #pragma unroll 1
    for (int k0 = 0; k0 < KDIM; k0 += 2 * BK) {
        if (k0 + BK < KDIM) stage(1, k0 + BK);
        compute(0);
        wait_stage();
        if (k0 + 2 * BK < KDIM) stage(0, k0 + 2 * BK);
        compute(1);
        wait_stage();
    }

    // ---- store 64x64 wave tile; C layout: VGPR r -> M = r + 8*kh, lane%16 -> N ----
    #pragma unroll
    for (int mi = 0; mi < 4; ++mi) {
        #pragma unroll
        for (int r = 0; r < 8; ++r) {
            const long long grow = row0 + wm * 64 + mi * 16 + r + 8 * kh;
            if (grow < size) {
                float* orow = out + (off + grow) * NDIM + wn * 64 + lm;
                #pragma unroll
                for (int ni = 0; ni < 4; ++ni)
                    __builtin_nontemporal_store(acc[mi][ni][r], orow + ni * 16);
            }
        }
    }
}

extern "C" void kernel_launch(void* const* d_in, const int* in_sizes, int n_in,
                              void* d_out, int out_size, void* d_ws, size_t ws_size,
                              hipStream_t stream) {
    (void)in_sizes; (void)n_in; (void)out_size; (void)d_ws; (void)ws_size;
    const float*     A     = (const float*)d_in[0];
    const float*     B     = (const float*)d_in[1];
    const long long* sizes = (const long long*)d_in[2];
    const long long* offs  = (const long long*)d_in[3];
    // d_in[4] = batch_padded_offsets (unused)
    float* out = (float*)d_out;

    dim3 grid(GCNT * MTILES);   // 8192 blocks; past-the-end row tiles self-exit
    dim3 block(256);            // 8 wave32
    GROUPED_GEMM_wmma_f32<<<grid, block, 0, stream>>>(A, B, sizes, offs, out);
}